// ModelClass_45724221833594
// MI455X (gfx1250) — compile-verified
//
#include <hip/hip_runtime.h>

// Problem constants (match reference)
#define GNUM  64
#define PNUM  1024
#define DDIM  64
#define KNN_K 6
#define NNODE (GNUM * PNUM)

typedef float v2f __attribute__((ext_vector_type(2)));
typedef float v8f __attribute__((ext_vector_type(8)));

// CDNA5 native f32 WMMA: D(16x16,f32) = A(16x4,f32) * B(4x16,f32) + C
// args: (neg_a, A, neg_b, B, c_mod, C, reuse_a, reuse_b)
__device__ __forceinline__ v8f wmma_f32(v2f a, v2f b, v8f c) {
    return __builtin_amdgcn_wmma_f32_16x16x4_f32(false, a, false, b, (short)0, c,
                                                 false, false);
}

// ---------------------------------------------------------------------------
// Kernel 1: per-graph kNN (k=6) via WMMA Gram tiles + register top-6 scan.
// One block (8 waves) per graph. Whole graph staged in LDS (CDNA5: 320KB/WGP).
// LDS layout (floats): pts[1024*66] | sq[1024] | scan[8 waves][16*17]
// A tile is register-hoisted across the 64-column-tile loop; the top-6 scan
// uses all 32 lanes (lane l: cols 0-7 of row l, lane l+16: cols 8-15), with a
// per-query-tile merge through the wave-coherent LDS buffer (DS ops are
// in-order within a wave on CDNA5, so no workgroup barrier is needed).
// ---------------------------------------------------------------------------
__global__ __launch_bounds__(256) void knn_kernel(const float* __restrict__ x,
                                                  int* __restrict__ nbr) {
    extern __shared__ float lds[];
    float* pts = lds;                      // padded row stride 66
    float* sq  = lds + 1024 * 66;
    const int g = blockIdx.x;
    const float* xg = x + (size_t)g * PNUM * DDIM;
    const int t = threadIdx.x;

    // Cooperative load + squared norms
    for (int r = t; r < PNUM; r += 256) {
        float ss = 0.f;
        const float4* src = (const float4*)(xg + (size_t)r * DDIM);
#pragma unroll
        for (int q = 0; q < 16; ++q) {
            float4 v = src[q];
            ss += v.x * v.x + v.y * v.y + v.z * v.z + v.w * v.w;
            float* dst = pts + r * 66 + q * 4;
            dst[0] = v.x; dst[1] = v.y; dst[2] = v.z; dst[3] = v.w;
        }
        sq[r] = ss;
    }
    __syncthreads();

    const int wave = t >> 5;
    const int lane = t & 31;
    const int lrow = lane & 15;
    const int kk   = (lane >> 4) << 1;   // WMMA K offset (0 or 2) per half-wave
    const int colh = (lane >> 4) * 8;    // scan column half (0 or 8)
    float* buf = lds + 1024 * 66 + 1024 + wave * (16 * 17);

    for (int qt = wave; qt < PNUM / 16; qt += 8) {
        const int qbase = qt * 16;

        // Hoist the invariant A tile (16 queries x 64 dims) into registers.
        v2f areg[16];
#pragma unroll
        for (int k0 = 0; k0 < DDIM; k0 += 4)
            areg[k0 >> 2] = *(const v2f*)(pts + (qbase + lrow) * 66 + k0 + kk);

        float bd[KNN_K];
        int   bi[KNN_K];
#pragma unroll
        for (int j = 0; j < KNN_K; ++j) { bd[j] = 3.0e38f; bi[j] = 0; }
        const float sqr  = sq[qbase + lrow];
        const int growl  = qbase + lrow;   // local index of this lane's query row

        for (int ct = 0; ct < PNUM / 16; ++ct) {
            const int cbase = ct * 16;
            v8f c = {};
#pragma unroll
            for (int k0 = 0; k0 < DDIM; k0 += 4) {
                v2f b = *(const v2f*)(pts + (cbase + lrow) * 66 + k0 + kk);
                c = wmma_f32(areg[k0 >> 2], b, c);
            }
            // Spill C tile (value(m=r+[lane>=16]*8, n=lane&15)) to wave scan buf.
#pragma unroll
            for (int r = 0; r < 8; ++r) {
                int row = r + ((lane >= 16) ? 8 : 0);
                buf[row * 17 + lrow] = c[r];
            }
            __builtin_amdgcn_wave_barrier();
            // All 32 lanes scan: row = lrow, columns colh..colh+7.
#pragma unroll
            for (int cc = 0; cc < 8; ++cc) {
                int col  = colh + cc;
                int gcol = cbase + col;
                float d2 = sqr + sq[gcol] - 2.0f * buf[lrow * 17 + col];
                if (gcol != growl && d2 < bd[KNN_K - 1]) {
                    bd[KNN_K - 1] = d2; bi[KNN_K - 1] = gcol;
#pragma unroll
                    for (int j = KNN_K - 1; j >= 1; --j) {
                        if (bd[j] < bd[j - 1]) {
                            float td = bd[j]; bd[j] = bd[j - 1]; bd[j - 1] = td;
                            int   ti = bi[j]; bi[j] = bi[j - 1]; bi[j - 1] = ti;
                        }
                    }
                }
            }
            __builtin_amdgcn_wave_barrier();
        }
        // Merge the two half-lists per row (disjoint column sets, no dups).
        if (lane >= 16) {
#pragma unroll
            for (int j = 0; j < KNN_K; ++j) {
                buf[lrow * 17 + j]          = bd[j];
                buf[lrow * 17 + KNN_K + j]  = __int_as_float(bi[j]);
            }
        }
        __builtin_amdgcn_wave_barrier();
        if (lane < 16) {
#pragma unroll
            for (int j = 0; j < KNN_K; ++j) {
                float d2 = buf[lrow * 17 + j];
                int   ii = __float_as_int(buf[lrow * 17 + KNN_K + j]);
                if (d2 < bd[KNN_K - 1]) {
                    bd[KNN_K - 1] = d2; bi[KNN_K - 1] = ii;
#pragma unroll
                    for (int jj = KNN_K - 1; jj >= 1; --jj) {
                        if (bd[jj] < bd[jj - 1]) {
                            float td = bd[jj]; bd[jj] = bd[jj - 1]; bd[jj - 1] = td;
                            int   ti = bi[jj]; bi[jj] = bi[jj - 1]; bi[jj - 1] = ti;
                        }
                    }
                }
            }
            size_t grow = (size_t)g * PNUM + qbase + lrow;
#pragma unroll
            for (int j = 0; j < KNN_K; ++j)
                nbr[grow * KNN_K + j] = g * PNUM + bi[j];  // global node index
        }
        __builtin_amdgcn_wave_barrier();
    }
}

// ---------------------------------------------------------------------------
// Kernel 2: N x 64 @ 64x64 GEMM + bias + PReLU (pre_nn layers).
// Block = 256 threads (8 waves), 128 rows/block. Optionally emits per-block
// (sum, sumsq) partials per feature for deterministic BN stats.
// ---------------------------------------------------------------------------
__global__ __launch_bounds__(256) void pre_gemm_kernel(
    const float* __restrict__ xin, const float* __restrict__ W,
    const float* __restrict__ bias, const float* __restrict__ alpha,
    float* __restrict__ out, float* __restrict__ partials, int doPart) {
    __shared__ float Wt[DDIM * 66];      // W transposed: Wt[n*66 + k]
    __shared__ float part[8 * 32 * 8];   // [wave][lane][ci*2 + {sum,sumsq}]
    const int t = threadIdx.x;
#pragma unroll
    for (int i = 0; i < 16; ++i) {
        int idx = t + 256 * i;
        Wt[(idx & 63) * 66 + (idx >> 6)] = W[idx];
    }
    __syncthreads();

    const int wave = t >> 5, lane = t & 31;
    const int lrow = lane & 15, kk = (lane >> 4) << 1, mofs = (lane >> 4) << 3;
    const int rowbase = blockIdx.x * 128 + wave * 16;

    v8f c0 = {}, c1 = {}, c2 = {}, c3 = {};
#pragma unroll
    for (int k0 = 0; k0 < DDIM; k0 += 4) {
        v2f a  = *(const v2f*)(xin + (size_t)(rowbase + lrow) * DDIM + k0 + kk);
        v2f b0 = *(const v2f*)(Wt + (0 * 16 + lrow) * 66 + k0 + kk);
        v2f b1 = *(const v2f*)(Wt + (1 * 16 + lrow) * 66 + k0 + kk);
        v2f b2 = *(const v2f*)(Wt + (2 * 16 + lrow) * 66 + k0 + kk);
        v2f b3 = *(const v2f*)(Wt + (3 * 16 + lrow) * 66 + k0 + kk);
        c0 = wmma_f32(a, b0, c0); c1 = wmma_f32(a, b1, c1);
        c2 = wmma_f32(a, b2, c2); c3 = wmma_f32(a, b3, c3);
    }
    v8f cc[4] = {c0, c1, c2, c3};
#pragma unroll
    for (int ci = 0; ci < 4; ++ci) {
        int n = ci * 16 + lrow;
        float bn = bias[n], an = alpha[n];
        float s1 = 0.f, s2 = 0.f;
#pragma unroll
        for (int r = 0; r < 8; ++r) {
            int row = rowbase + mofs + r;
            float v = cc[ci][r] + bn;
            v = (v >= 0.f) ? v : an * v;
            out[(size_t)row * DDIM + n] = v;
            s1 += v; s2 += v * v;
        }
        if (doPart) {
            part[(wave * 32 + lane) * 8 + ci * 2 + 0] = s1;
            part[(wave * 32 + lane) * 8 + ci * 2 + 1] = s2;
        }
    }
    if (doPart) {
        __syncthreads();
        if (t < 64) {
            int n = t, ci = n >> 4, l = n & 15;
            float s1 = 0.f, s2 = 0.f;
#pragma unroll
            for (int w = 0; w < 8; ++w) {
                s1 += part[(w * 32 + l) * 8 + ci * 2] + part[(w * 32 + 16 + l) * 8 + ci * 2];
                s2 += part[(w * 32 + l) * 8 + ci * 2 + 1] + part[(w * 32 + 16 + l) * 8 + ci * 2 + 1];
            }
            partials[(size_t)blockIdx.x * 128 + n]      = s1;
            partials[(size_t)blockIdx.x * 128 + 64 + n] = s2;
        }
    }
}

// Deterministic BN stat reduction -> scale/shift (fixed iteration order)
__global__ void bn_reduce_kernel(const float* __restrict__ partials, int nblocks,
                                 const float* __restrict__ gamma,
                                 const float* __restrict__ beta,
                                 float* __restrict__ bnp) {
    int n = threadIdx.x;
    if (n < 64) {
        float s1 = 0.f, s2 = 0.f;
        for (int b = 0; b < nblocks; ++b) {
            s1 += partials[(size_t)b * 128 + n];
            s2 += partials[(size_t)b * 128 + 64 + n];
        }
        float inv = 1.0f / (float)NNODE;
        float mu  = s1 * inv;
        float var = s2 * inv - mu * mu;
        float sc  = rsqrtf(var + 1e-5f) * gamma[n];
        bnp[n] = sc;
        bnp[64 + n] = beta[n] - mu * sc;
    }
}

__global__ __launch_bounds__(256) void bn_apply_kernel(
    const float* __restrict__ hin, const float* __restrict__ bnp,
    float* __restrict__ hout) {
    int idx = blockIdx.x * 256 + threadIdx.x;  // float4 index
    int d0  = (idx * 4) & 63;
    float4 v  = ((const float4*)hin)[idx];
    float4 sc = *(const float4*)(bnp + d0);
    float4 sh = *(const float4*)(bnp + 64 + d0);
    float4 o;
    o.x = v.x * sc.x + sh.x; o.y = v.y * sc.y + sh.y;
    o.z = v.z * sc.z + sh.z; o.w = v.w * sc.w + sh.w;
    ((float4*)hout)[idx] = o;
}

// ---------------------------------------------------------------------------
// Kernel 3: GeneralConv layer, fused via linearity:
//   h_out = PReLU( (sum_{k} h[nbr[i][k]]) @ W  +  K*b  +  h[i] , act_a )
// Phase 1 gathers 6-neighbor sums into LDS, phase 2 does the WMMA GEMM.
// ---------------------------------------------------------------------------
__global__ __launch_bounds__(256) void conv_kernel(
    const float* __restrict__ hin, const int* __restrict__ nbr,
    const float* __restrict__ W, const float* __restrict__ bias,
    const float* __restrict__ alpha, float* __restrict__ hout) {
    __shared__ float Wt[DDIM * 66];
    __shared__ float sA[128 * 66];
    const int t = threadIdx.x;
#pragma unroll
    for (int i = 0; i < 16; ++i) {
        int idx = t + 256 * i;
        Wt[(idx & 63) * 66 + (idx >> 6)] = W[idx];
    }
    {   // gather-sum: 2 threads per row, 32 dims each
        int rl = t >> 1, half = t & 1, d0 = half * 32;
        size_t grow = (size_t)blockIdx.x * 128 + rl;
        const int* nb = nbr + grow * KNN_K;
        float acc[32];
#pragma unroll
        for (int q = 0; q < 32; ++q) acc[q] = 0.f;
        for (int j = 0; j < KNN_K; ++j) {
            if (j + 1 < KNN_K)  // CDNA5 global_prefetch_b8 for next gather row
                __builtin_prefetch(hin + (size_t)nb[j + 1] * DDIM + d0, 0, 1);
            const float* hp = hin + (size_t)nb[j] * DDIM + d0;
#pragma unroll
            for (int q = 0; q < 8; ++q) {
                float4 v = *(const float4*)(hp + q * 4);
                acc[q * 4 + 0] += v.x; acc[q * 4 + 1] += v.y;
                acc[q * 4 + 2] += v.z; acc[q * 4 + 3] += v.w;
            }
        }
#pragma unroll
        for (int q = 0; q < 32; ++q) sA[rl * 66 + d0 + q] = acc[q];
    }
    __syncthreads();

    const int wave = t >> 5, lane = t & 31;
    const int lrow = lane & 15, kk = (lane >> 4) << 1, mofs = (lane >> 4) << 3;
    const int rowbase = blockIdx.x * 128 + wave * 16;
    v8f c0 = {}, c1 = {}, c2 = {}, c3 = {};
#pragma unroll
    for (int k0 = 0; k0 < DDIM; k0 += 4) {
        v2f a  = *(const v2f*)(sA + (wave * 16 + lrow) * 66 + k0 + kk);
        v2f b0 = *(const v2f*)(Wt + (0 * 16 + lrow) * 66 + k0 + kk);
        v2f b1 = *(const v2f*)(Wt + (1 * 16 + lrow) * 66 + k0 + kk);
        v2f b2 = *(const v2f*)(Wt + (2 * 16 + lrow) * 66 + k0 + kk);
        v2f b3 = *(const v2f*)(Wt + (3 * 16 + lrow) * 66 + k0 + kk);
        c0 = wmma_f32(a, b0, c0); c1 = wmma_f32(a, b1, c1);
        c2 = wmma_f32(a, b2, c2); c3 = wmma_f32(a, b3, c3);
    }
    v8f cc[4] = {c0, c1, c2, c3};
#pragma unroll
    for (int ci = 0; ci < 4; ++ci) {
        int n = ci * 16 + lrow;
        float bn = (float)KNN_K * bias[n], an = alpha[n];
#pragma unroll
        for (int r = 0; r < 8; ++r) {
            int row = rowbase + mofs + r;
            float v = cc[ci][r] + bn + hin[(size_t)row * DDIM + n];
            v = (v >= 0.f) ? v : an * v;
            hout[(size_t)row * DDIM + n] = v;
        }
    }
}

// Deterministic per-graph pooling of one 64-wide layer into pooled[g][320]
__global__ __launch_bounds__(256) void pool_kernel(const float* __restrict__ h,
                                                   float* __restrict__ pooled,
                                                   int layer) {
    __shared__ float red[4 * 64];
    int g = blockIdx.x;
    int n = threadIdx.x & 63, chunk = threadIdx.x >> 6;
    float s = 0.f;
    const float* base = h + ((size_t)g * PNUM + chunk * 256) * DDIM + n;
    for (int r = 0; r < 256; ++r) s += base[(size_t)r * DDIM];
    red[chunk * 64 + n] = s;
    __syncthreads();
    if (threadIdx.x < 64) {
        int nn = threadIdx.x;
        pooled[g * 320 + layer * 64 + nn] =
            red[nn] + red[64 + nn] + red[128 + nn] + red[192 + nn];
    }
}

// ---------------------------------------------------------------------------
// Kernel 4: FFN  [64,320] @ [320,320] + b1 -> LeakyReLU -> @ [320,1] + b2
// Single block, 8 waves, 80 WMMA output tiles; z1 staged in LDS (stride 321).
// ---------------------------------------------------------------------------
__global__ __launch_bounds__(256) void ffn_kernel(
    const float* __restrict__ pooled, const float* __restrict__ W1,
    const float* __restrict__ b1, const float* __restrict__ W2,
    const float* __restrict__ b2, float* __restrict__ out) {
    extern __shared__ float zb[];  // 64 * 321 floats
    const int t = threadIdx.x, wave = t >> 5, lane = t & 31;
    const int lrow = lane & 15, kk = (lane >> 4) << 1, mofs = (lane >> 4) << 3;

    for (int tid = wave; tid < 80; tid += 8) {  // uniform: 10 tiles per wave
        int rt = tid / 20, ctile = tid % 20;
        int n = ctile * 16 + lrow;
        v8f c = {};
        for (int k0 = 0; k0 < 320; k0 += 4) {
            v2f a = *(const v2f*)(pooled + (size_t)(rt * 16 + lrow) * 320 + k0 + kk);
            v2f b;
            b.x = W1[(size_t)(k0 + kk) * 320 + n];
            b.y = W1[(size_t)(k0 + kk + 1) * 320 + n];
            c = wmma_f32(a, b, c);
        }
        float bn = b1[n];
#pragma unroll
        for (int r = 0; r < 8; ++r) {
            int row = rt * 16 + mofs + r;
            float v = c[r] + bn;
            v = (v >= 0.f) ? v : 0.01f * v;
            zb[row * 321 + n] = v;
        }
    }
    __syncthreads();
    if (t < 64) {
        float acc = b2[0];
        for (int k = 0; k < 320; ++k) acc += zb[t * 321 + k] * W2[k];
        out[t] = acc;
    }
}

// ---------------------------------------------------------------------------
extern "C" void kernel_launch(void* const* d_in, const int* in_sizes, int n_in,
                              void* d_out, int out_size, void* d_ws,
                              size_t ws_size, hipStream_t stream) {
    (void)in_sizes; (void)n_in; (void)out_size; (void)ws_size;
    const float* x     = (const float*)d_in[0];
    // d_in[1] batchidxs: layout is known (P consecutive nodes per graph) -> unused
    const float* preW1 = (const float*)d_in[2];
    const float* preb1 = (const float*)d_in[3];
    const float* prea1 = (const float*)d_in[4];
    const float* preW2 = (const float*)d_in[5];
    const float* preb2 = (const float*)d_in[6];
    const float* prea2 = (const float*)d_in[7];
    const float* gamma = (const float*)d_in[8];
    const float* beta  = (const float*)d_in[9];
    const float* acta  = (const float*)d_in[10];
    const float* convW = (const float*)d_in[11];
    const float* convb = (const float*)d_in[12];
    const float* ffnW1 = (const float*)d_in[13];
    const float* ffnb1 = (const float*)d_in[14];
    const float* ffnW2 = (const float*)d_in[15];
    const float* ffnb2 = (const float*)d_in[16];
    float* out = (float*)d_out;

    // Workspace carve-up (~35.1 MB total)
    char* ws = (char*)d_ws;
    int*   nbr      = (int*)(ws);                                  // 1.5 MB
    float* hA       = (float*)(ws + (size_t)2 * 1024 * 1024);      // 16 MB
    float* hB       = (float*)(ws + (size_t)18 * 1024 * 1024);     // 16 MB
    float* partials = (float*)(ws + (size_t)34 * 1024 * 1024);     // 256 KB
    float* bnp      = (float*)(ws + (size_t)34 * 1024 * 1024 + 512 * 1024);
    float* pooled   = (float*)(ws + (size_t)35 * 1024 * 1024);     // 80 KB

    const int nGemmBlk = NNODE / 128;  // 512

    size_t knn_lds = (size_t)(1024 * 66 + 1024 + 8 * 16 * 17) * sizeof(float);
    knn_kernel<<<GNUM, 256, knn_lds, stream>>>(x, nbr);

    pre_gemm_kernel<<<nGemmBlk, 256, 0, stream>>>(x, preW1, preb1, prea1, hA,
                                                  partials, 0);
    pre_gemm_kernel<<<nGemmBlk, 256, 0, stream>>>(hA, preW2, preb2, prea2, hB,
                                                  partials, 1);
    bn_reduce_kernel<<<1, 64, 0, stream>>>(partials, nGemmBlk, gamma, beta, bnp);
    bn_apply_kernel<<<(NNODE * DDIM / 4) / 256, 256, 0, stream>>>(hB, bnp, hA);
    pool_kernel<<<GNUM, 256, 0, stream>>>(hA, pooled, 0);

    float* hin = hA;
    float* hout = hB;
    for (int i = 0; i < 4; ++i) {
        conv_kernel<<<nGemmBlk, 256, 0, stream>>>(hin, nbr,
                                                  convW + (size_t)i * DDIM * DDIM,
                                                  convb + (size_t)i * DDIM, acta,
                                                  hout);
        pool_kernel<<<GNUM, 256, 0, stream>>>(hout, pooled, i + 1);
        float* tmp = hout; hout = hin; hin = tmp;
    }

    ffn_kernel<<<1, 256, (size_t)64 * 321 * sizeof(float), stream>>>(
        pooled, ffnW1, ffnb1, ffnW2, ffnb2, out);
}